// Sparsemax_21363167330753
// MI455X (gfx1250) — compile-verified
//
#include <hip/hip_runtime.h>
#include <stdint.h>

// Sparsemax over rows of length D=4096, one row per block (16384 blocks).
// Bandwidth-bound target: 512 MiB total traffic -> ~22us floor at 23.3 TB/s.
// CDNA5 path: global_load_async_to_lds_b128 staging + s_wait_asynccnt.
// VALU kept minimal: tau search runs on a tiny candidate set (z > max-1)
// solved by wave 0 with shuffle-only reductions; only 5 block barriers total.

#define D   4096
#define T   256          // threads per block (8 wave32)
#define E   (D / T)      // 16 elements per thread
#define NW  (T / 32)     // 8 waves

__device__ __forceinline__ float wave_sum_f(float v) {
#pragma unroll
  for (int m = 16; m >= 1; m >>= 1) v += __shfl_xor(v, m, 32);
  return v;
}

__device__ __forceinline__ float wave_max_f(float v) {
#pragma unroll
  for (int m = 16; m >= 1; m >>= 1) v = fmaxf(v, __shfl_xor(v, m, 32));
  return v;
}

__global__ __launch_bounds__(T) void sparsemax_kernel(const float* __restrict__ x,
                                                      float* __restrict__ out,
                                                      int rows) {
  // s_row: async staging buffer, then reused as the candidate buffer.
  __shared__ __attribute__((aligned(16))) float s_row[D];
  __shared__ float s_fscr[NW];   // wave maxima
  __shared__ int   s_iscr[NW];   // wave candidate counts
  __shared__ float s_tau;        // solved threshold (shifted domain)

  const int row = blockIdx.x;
  if (row >= rows) return;                 // uniform per block; barriers safe
  const int tid  = threadIdx.x;
  const int lane = tid & 31;
  const int wid  = tid >> 5;
  const float* g = x + (size_t)row * D;

  // ---- Stage row into LDS via CDNA5 async path (4 x b128 per lane) -------
  const uint32_t lds0 = (uint32_t)(uintptr_t)&s_row[0];  // flat low 32b == LDS offset
#pragma unroll
  for (int i = 0; i < (D * 4) / (T * 16); ++i) {         // 4 iterations
    const uint32_t off = (uint32_t)((i * T + tid) * 16);
    asm volatile("global_load_async_to_lds_b128 %0, %1, off"
                 :
                 : "v"(lds0 + off), "v"((const void*)((const char*)g + off))
                 : "memory");
  }
  asm volatile("s_wait_asynccnt 0x0" ::: "memory");
  __syncthreads();                                       // barrier 1

  // ---- Row -> registers (conflict-free strided layout), local max --------
  float z[E];
  float mx = -INFINITY;
#pragma unroll
  for (int j = 0; j < E; ++j) {
    z[j] = s_row[j * T + tid];
    mx = fmaxf(mx, z[j]);
  }

  // ---- Block max (deterministic: fixed shuffle tree + ordered LDS max) ---
  mx = wave_max_f(mx);
  if (lane == 0) s_fscr[wid] = mx;
  __syncthreads();                                       // barrier 2
  mx = s_fscr[0];
#pragma unroll
  for (int i = 1; i < NW; ++i) mx = fmaxf(mx, s_fscr[i]);

  // ---- Candidate filter: only z > mx-1 can be in the support -------------
  const float thresh = mx - 1.0f;
  int cnt = 0;
#pragma unroll
  for (int j = 0; j < E; ++j) cnt += (z[j] > thresh) ? 1 : 0;

  // Exclusive block scan of cnt (shuffle scan + wave totals), deterministic.
  int inc = cnt;
#pragma unroll
  for (int m = 1; m < 32; m <<= 1) {
    int t = __shfl_up(inc, m, 32);
    if (lane >= m) inc += t;
  }
  if (lane == 31) s_iscr[wid] = inc;       // wave total
  __syncthreads();                                       // barrier 3
  int waveBase = 0, C = 0;
#pragma unroll
  for (int i = 0; i < NW; ++i) {
    waveBase += (i < wid) ? s_iscr[i] : 0;
    C += s_iscr[i];
  }
  int pos = waveBase + (inc - cnt);        // exclusive prefix for this thread

  // Compact shifted candidates into LDS (staging buffer is dead now).
#pragma unroll
  for (int j = 0; j < E; ++j) {
    if (z[j] > thresh) s_row[pos++] = z[j] - mx;
  }
  __syncthreads();                                       // barrier 4

  // ---- Wave 0 solves tau on the candidate set (shuffle-only, no barriers)
  if (wid == 0) {
    // Bisection: f(tau) = sum(relu(zc - tau)) is monotone; tau* in [-1, 0).
    float lo = -1.0f, hi = 0.0f;
#pragma unroll 1
    for (int it = 0; it < 26; ++it) {
      const float tau = 0.5f * (lo + hi);
      float p = 0.0f;
      for (int i = lane; i < C; i += 32) p += fmaxf(s_row[i] - tau, 0.0f);
      const float f = wave_sum_f(p);
      if (f >= 1.0f) lo = tau; else hi = tau;            // lane-uniform
    }
    // Michelot refinement: exact tau on the final linear segment.
    float tau = lo;
#pragma unroll 1
    for (int it = 0; it < 3; ++it) {
      float s = 0.0f, k = 0.0f;
      for (int i = lane; i < C; i += 32) {
        const float v = s_row[i];
        if (v > tau) { s += v; k += 1.0f; }
      }
      s = wave_sum_f(s);
      k = wave_sum_f(k);
      tau = (s - 1.0f) / fmaxf(k, 1.0f);
    }
    if (lane == 0) s_tau = tau;
  }
  __syncthreads();                                       // barrier 5

  // ---- Output: relu((z - mx) - tau), coalesced 128B/wave b32 stores ------
  const float tau = s_tau;
  float* o = out + (size_t)row * D;
#pragma unroll
  for (int j = 0; j < E; ++j) o[j * T + tid] = fmaxf((z[j] - mx) - tau, 0.0f);
}

extern "C" void kernel_launch(void* const* d_in, const int* in_sizes, int n_in,
                              void* d_out, int out_size, void* d_ws, size_t ws_size,
                              hipStream_t stream) {
  (void)n_in; (void)d_ws; (void)ws_size; (void)out_size;
  const float* x = (const float*)d_in[0];
  float* out = (float*)d_out;
  const int rows = in_sizes[0] / D;        // 16384 for (4, 4096, 4096)
  sparsemax_kernel<<<rows, T, 0, stream>>>(x, out, rows);
}